// NMS_v0_12335146074350
// MI455X (gfx1250) — compile-verified
//
#include <hip/hip_runtime.h>
#include <stdint.h>

#define TOPK  1000
#define NBINS 65536
#define CAP   4096   // candidate capacity (expected ~1000-1100)

typedef float v4f __attribute__((ext_vector_type(4)));

// ---------------------------------------------------------------------------
// ws layout (bytes):
//   keys : [0, 4*M)                       per-row order key (flip-float)
//   hist : [4*M, 4*M+4*NBINS)             16-bit-key histogram
//   meta : next 16 uints                  [0]=n_cand (atomic), [1]=threshold bin
//   cand : next CAP*8 bytes (u64, 8-al.)  (key<<32)|idx composite records
// ---------------------------------------------------------------------------

__device__ __forceinline__ unsigned f2key(float f) {
  unsigned u = __float_as_uint(f);
  return (u & 0x80000000u) ? ~u : (u | 0x80000000u);  // monotone float->uint
}

// K0: zero output + histogram + meta (harness poisons, and we must rewrite
// every element of d_out each call).
__global__ void k_zero(int* __restrict__ out, int out_n,
                       unsigned* __restrict__ hist_meta, int hm_n) {
  int i = blockIdx.x * blockDim.x + threadIdx.x;
  if (i < out_n) out[i] = 0;
  if (i < hm_n) hist_meta[i] = 0u;
}

// K1: per-row max of 128 f32 (one wave32 per row; 512B coalesced via b128 NT
// loads -- data is read-once, keep it out of the way of L2), then key +
// histogram. This kernel moves 128MB and bounds the whole problem.
__global__ void k_score_hist(const float* __restrict__ logits,
                             unsigned* __restrict__ keys,
                             unsigned* __restrict__ hist, int M) {
  int wave = threadIdx.x >> 5;
  int lane = threadIdx.x & 31;
  int row = blockIdx.x * 8 + wave;
  if (row >= M) return;
  const v4f* p =
      reinterpret_cast<const v4f*>(logits + (size_t)row * 128 + lane * 4);
  v4f v = __builtin_nontemporal_load(p);
  float m = fmaxf(fmaxf(v.x, v.y), fmaxf(v.z, v.w));
#pragma unroll
  for (int off = 16; off > 0; off >>= 1)
    m = fmaxf(m, __shfl_xor(m, off, 32));
  if (lane == 0) {
    unsigned key = f2key(m);
    keys[row] = key;
    atomicAdd(&hist[key >> 16], 1u);
  }
}

// K2: find histogram bin T containing rank TOPK (descending).
__global__ void k_threshold(const unsigned* __restrict__ hist,
                            unsigned* __restrict__ meta) {
  __shared__ unsigned partial[1024];
  int t = threadIdx.x;
  unsigned s = 0;
  for (int b = 0; b < NBINS / 1024; ++b) s += hist[t * (NBINS / 1024) + b];
  partial[t] = s;
  __syncthreads();
  if (t == 0) {
    unsigned acc = 0;
    int chunk = 0;
    for (int c = 1023; c >= 0; --c) {
      if (acc + partial[c] >= TOPK) { chunk = c; break; }
      acc += partial[c];
    }
    unsigned T = 0;
    for (int b = chunk * 64 + 63; b >= chunk * 64; --b) {
      unsigned h = hist[b];
      if (acc + h >= TOPK) { T = (unsigned)b; break; }
      acc += h;
    }
    meta[1] = T;
  }
}

// K3: gather candidates with key16 >= T as (key<<32)|idx. Sorting this
// composite descending matches the reference tie-break exactly (equal score
// -> larger original index first), and makes the result order-deterministic.
__global__ void k_gather(const unsigned* __restrict__ keys,
                         unsigned* __restrict__ meta,
                         unsigned long long* __restrict__ cand, int M) {
  int m = blockIdx.x * blockDim.x + threadIdx.x;
  if (m >= M) return;
  unsigned key = keys[m];
  unsigned T = meta[1];
  if ((key >> 16) >= T) {
    unsigned pos = atomicAdd(&meta[0], 1u);
    if (pos < CAP)
      cand[pos] = ((unsigned long long)key << 32) | (unsigned)m;
  }
}

// K4: single block -- bitonic sort candidates (desc), async-gather the
// top-K boxes into LDS with the CDNA5 async data mover, then wave32
// ballot-driven greedy NMS (i suppressed iff it overlaps an earlier-kept
// interval, exact for OVERLAP==0 since iou>0 <=> inter>0).
__global__ __launch_bounds__(1024) void k_sort_nms(
    const unsigned long long* __restrict__ cand_g,
    const unsigned* __restrict__ meta,
    const float* __restrict__ boxes,  // (M,2) = (x1,x2)
    int* __restrict__ out, int M) {
  __shared__ unsigned long long cand_s[CAP];  // 32KB
  __shared__ float2 sbox[1024];               // gathered (x1,x2), sorted order
  __shared__ float kx1[1024];
  __shared__ float kx2[1024];

  int tid = threadIdx.x;
  unsigned n = meta[0];
  if (n > CAP) n = CAP;
  unsigned N = 1024;                 // n >= TOPK by construction
  while (N < n) N <<= 1;             // 1024 / 2048 / 4096

  for (unsigned i = tid; i < N; i += 1024)
    cand_s[i] = (i < n) ? cand_g[i] : 0ull;  // 0 sorts last
  __syncthreads();

  // bitonic sort, descending
  for (unsigned k = 2; k <= N; k <<= 1) {
    for (unsigned j = k >> 1; j > 0; j >>= 1) {
      for (unsigned i = tid; i < N; i += 1024) {
        unsigned ixj = i ^ j;
        if (ixj > i) {
          unsigned long long a = cand_s[i], b = cand_s[ixj];
          bool desc = ((i & k) == 0);
          if (desc ? (a < b) : (a > b)) { cand_s[i] = b; cand_s[ixj] = a; }
        }
      }
      __syncthreads();
    }
  }

  int Kp = (int)((n < (unsigned)TOPK) ? n : (unsigned)TOPK);

  // Async indexed gather: each active lane DMAs its (x1,x2) pair straight
  // into LDS (GLOBAL_LOAD_ASYNC_TO_LDS_B64, GVS mode, tracked by ASYNCcnt).
  if (tid < Kp) {
    unsigned idx = (unsigned)cand_s[tid];            // original flat index
    unsigned lds_addr = (unsigned)(uintptr_t)(&sbox[tid]);
    unsigned voff = idx * 8u;                        // byte offset into boxes
    unsigned long long base = (unsigned long long)(uintptr_t)boxes;
    asm volatile("global_load_async_to_lds_b64 %0, %1, %2"
                 :: "v"(lds_addr), "v"(voff), "s"(base)
                 : "memory");
  }
  asm volatile("s_wait_asynccnt 0" ::: "memory");
  __syncthreads();

  // Wave-0 greedy NMS: candidate alive iff it overlaps no earlier-kept
  // interval; intra-chunk order resolved with ballot + ffs serialization.
  if (tid < 32) {
    int lane = tid;
    int kc = 0;
    int nChunks = (Kp + 31) >> 5;
    for (int c = 0; c < nChunks; ++c) {
      int j = c * 32 + lane;
      bool valid = (j < Kp);
      int jj = valid ? j : 0;
      float x1 = sbox[jj].x;
      float x2 = sbox[jj].y;
      int myidx = (int)(unsigned)cand_s[jj];
      bool alive = valid;
      for (int t = 0; t < kc; ++t) {
        float inter = fminf(x2, kx2[t]) - fmaxf(x1, kx1[t]);
        alive = alive && !(inter > 0.0f);
      }
      unsigned long long mask = __ballot(alive ? 1 : 0);
      while (mask) {
        int b = __ffsll(mask) - 1;                 // next kept in order
        float bx1 = __shfl(x1, b, 32);
        float bx2 = __shfl(x2, b, 32);
        int bidx = __shfl(myidx, b, 32);
        if (lane == 0) {
          kx1[kc] = bx1;
          kx2[kc] = bx2;
          out[M + kc] = bidx;                      // row 1 of (2,M) output
        }
        kc++;
        float inter = fminf(x2, bx2) - fmaxf(x1, bx1);
        alive = alive && (lane > b) && !(inter > 0.0f);
        mask = __ballot(alive ? 1 : 0);
      }
    }
  }
}

extern "C" void kernel_launch(void* const* d_in, const int* in_sizes, int n_in,
                              void* d_out, int out_size, void* d_ws,
                              size_t ws_size, hipStream_t stream) {
  const float* logits = (const float*)d_in[0];   // (16,16384,128) f32
  const float* boxes  = (const float*)d_in[1];   // (16,16384,2)   f32, sorted pairs
  int* out = (int*)d_out;                        // (2, M) int32
  int M = in_sizes[0] / 128;                     // 262144

  char* ws = (char*)d_ws;
  unsigned* keys = (unsigned*)ws;
  unsigned* hist = (unsigned*)(ws + (size_t)M * 4);
  unsigned* meta = hist + NBINS;
  unsigned long long* cand = (unsigned long long*)(meta + 16);

  k_zero<<<(out_size + 255) / 256, 256, 0, stream>>>(out, out_size, hist,
                                                     NBINS + 16);
  k_score_hist<<<(M + 7) / 8, 256, 0, stream>>>(logits, keys, hist, M);
  k_threshold<<<1, 1024, 0, stream>>>(hist, meta);
  k_gather<<<(M + 255) / 256, 256, 0, stream>>>(keys, meta, cand, M);
  k_sort_nms<<<1, 1024, 0, stream>>>(cand, meta, boxes, out, M);
}